// Sub_MGU_27582279975062
// MI455X (gfx1250) — compile-verified
//
#include <hip/hip_runtime.h>
#include <hip/hip_bf16.h>

typedef float v2f __attribute__((ext_vector_type(2)));
typedef float v8f __attribute__((ext_vector_type(8)));

#define SUB_NO 32
#define HID    16
#define BATCH  64
#define TSTEPS 2048
#define OUTW   (SUB_NO * HID)   // 512

// One wave (32 lanes) per block: one subunit s x one 16-row batch tile.
// Canonical state layout = WMMA D-format for a 16x16 f32 tile:
//   element (m=batch_row, n=hid): vgpr r = m&7, lane = n + 16*(m>>3)
// A-format (matmul input, K-chunk j, vgpr v): element (m=lane&15, k=4j + v + 2*(lane>>4))
// B-format (weights Wt, K-chunk j, vgpr v):   element (k=4j + v + 2*(lane>>4), n=lane&15)
__global__ __launch_bounds__(32)
void sub_mgu_kernel(const float* __restrict__ x,       // [B, T, S]
                    const float* __restrict__ W_hif,   // [S, H]
                    const float* __restrict__ W_hin,   // [S, H]
                    const float* __restrict__ W_hhf,   // [S, H, H]
                    const float* __restrict__ W_hhn,   // [S, H, H]
                    const float* __restrict__ bias_hi, // [2*S*H]
                    const float* __restrict__ bias_hh, // [2*S*H]
                    float* __restrict__ out)           // [B, T, S*H]
{
    __shared__ float smem[2][HID * HID];  // ping-pong transpose buffers (2KB)

    const int lane = threadIdx.x & 31;
    const int nn   = lane & 15;     // hidden index (N / lane)
    const int hi   = lane >> 4;     // lane half
    const int s    = blockIdx.x & (SUB_NO - 1);
    const int bb   = (blockIdx.x >> 5) * 16;   // batch tile base

    // ---- Preload B operands: Wt[k,n] = W[s,n,k]; chunk j vgpr pair covers k = 4j+2hi, +1
    const float* wfrow = W_hhf + ((size_t)(s * HID + nn)) * HID;
    const float* wnrow = W_hhn + ((size_t)(s * HID + nn)) * HID;
    v2f bf[4], bn[4];
#pragma unroll
    for (int j = 0; j < 4; ++j) {
        bf[j] = *(const v2f*)(wfrow + 4 * j + 2 * hi);
        bn[j] = *(const v2f*)(wnrow + 4 * j + 2 * hi);
    }

    // ---- Per-lane (per-n) scalars
    const float w_if = W_hif[s * HID + nn];
    const float w_in = W_hin[s * HID + nn];
    const float b_if = bias_hi[s * HID + nn];
    const float b_in = bias_hi[SUB_NO * HID + s * HID + nn];
    const float b_hf = bias_hh[s * HID + nn];
    const float b_hn = bias_hh[SUB_NO * HID + s * HID + nn];

    // C seeds (D-format): bias depends only on n -> all components equal
    v8f Cf, Cn, Zv;
#pragma unroll
    for (int r = 0; r < 8; ++r) { Cf[r] = b_if + b_hf; Cn[r] = b_hn; Zv[r] = 0.0f; }

    // ---- State: hD (D-format), hA (A-format K-chunks). h0 = 0.
    float hD[8];
    v2f   hA[4];
#pragma unroll
    for (int r = 0; r < 8; ++r) hD[r] = 0.0f;
#pragma unroll
    for (int j = 0; j < 4; ++j) { hA[j][0] = 0.0f; hA[j][1] = 0.0f; }

    // x chunk loads: lane j reads x[bb + (j&15), tc + (j>>4) + 2p, s]
    //   -> one b32 load covers 2 timesteps; 4 loads cover an 8-step chunk.
    const float* const xpb =
        x + (size_t)(bb + nn) * TSTEPS * SUB_NO + (size_t)hi * SUB_NO + s;
    float xq[4], xqn[4];
#pragma unroll
    for (int p = 0; p < 4; ++p) xq[p] = xpb[2 * p * SUB_NO];   // chunk tc=0

    float* const obase = out + (size_t)s * HID + nn;
    const int idxbase = hi * 32;   // bpermute byte base: src lane = r + 8*hi + 16*dt

#pragma unroll 1
    for (int tc = 0; tc < TSTEPS; tc += 8) {
        // ---- prefetch next 8-step x chunk (clamped; redundant reload on last chunk)
        int tnext = tc + 8; if (tnext > TSTEPS - 8) tnext = TSTEPS - 8;
        const float* xnp = xpb + (size_t)tnext * SUB_NO;
#pragma unroll
        for (int p = 0; p < 4; ++p) xqn[p] = xnp[2 * p * SUB_NO];

#pragma unroll
        for (int u = 0; u < 8; ++u) {
            const int t  = tc + u;
            const int p  = u >> 1;      // which x register
            const int dt = u & 1;       // which half of its lanes
            const int pb = u & 1;       // ping-pong LDS buffer

            // broadcast per-batch scalar x to D-format rows
            float xb[8];
#pragma unroll
            for (int r = 0; r < 8; ++r)
                xb[r] = __int_as_float(__builtin_amdgcn_ds_bpermute(
                    idxbase + 4 * r + 64 * dt, __float_as_int(xq[p])));

            // ---- gate matmuls: two parallel 2-deep WMMA chains per gate
            v8f fa = __builtin_amdgcn_wmma_f32_16x16x4_f32(
                false, hA[0], false, bf[0], (short)0, Cf, false, false);
            v8f fb = __builtin_amdgcn_wmma_f32_16x16x4_f32(
                false, hA[2], false, bf[2], (short)0, Zv, false, false);
            v8f na = __builtin_amdgcn_wmma_f32_16x16x4_f32(
                false, hA[0], false, bn[0], (short)0, Cn, false, false);
            v8f nb = __builtin_amdgcn_wmma_f32_16x16x4_f32(
                false, hA[2], false, bn[2], (short)0, Zv, false, false);
            fa = __builtin_amdgcn_wmma_f32_16x16x4_f32(
                false, hA[1], false, bf[1], (short)0, fa, false, false);
            fb = __builtin_amdgcn_wmma_f32_16x16x4_f32(
                false, hA[3], false, bf[3], (short)0, fb, false, false);
            na = __builtin_amdgcn_wmma_f32_16x16x4_f32(
                false, hA[1], false, bn[1], (short)0, na, false, false);
            nb = __builtin_amdgcn_wmma_f32_16x16x4_f32(
                false, hA[3], false, bn[3], (short)0, nb, false, false);

            // ---- elementwise MGU update (D-format)
#pragma unroll
            for (int r = 0; r < 8; ++r) {
                const float zf = (fa[r] + fb[r]) + xb[r] * w_if;
                const float f  = __builtin_amdgcn_rcpf(1.0f + __expf(-zf));     // sigmoid
                const float zn = xb[r] * w_in + b_in + f * (na[r] + nb[r]);
                const float nt = 1.0f -
                    2.0f * __builtin_amdgcn_rcpf(__expf(2.0f * zn) + 1.0f);     // tanh
                hD[r] = f * nt + (1.0f - f) * hD[r];
            }

            // ---- stream h_new to ys (16 lanes x b32 = 64B contiguous per row)
#pragma unroll
            for (int r = 0; r < 8; ++r)
                obase[((size_t)(bb + r + 8 * hi) * TSTEPS + t) * OUTW] = hD[r];

            // ---- D-format -> A-format transpose (ping-pong LDS buffer)
#pragma unroll
            for (int r = 0; r < 8; ++r)
                smem[pb][(r + 8 * hi) * HID + nn] = hD[r];
            asm volatile("s_wait_dscnt 0" ::: "memory");
#pragma unroll
            for (int j = 0; j < 4; ++j)
                hA[j] = *(const v2f*)(&smem[pb][nn * HID + 4 * j + 2 * hi]);
        }

#pragma unroll
        for (int p = 0; p < 4; ++p) xq[p] = xqn[p];
    }
}

extern "C" void kernel_launch(void* const* d_in, const int* in_sizes, int n_in,
                              void* d_out, int out_size, void* d_ws, size_t ws_size,
                              hipStream_t stream) {
    const float* x       = (const float*)d_in[0];
    const float* W_hif   = (const float*)d_in[1];
    const float* W_hin   = (const float*)d_in[2];
    const float* W_hhf   = (const float*)d_in[3];
    const float* W_hhn   = (const float*)d_in[4];
    const float* bias_hi = (const float*)d_in[5];
    const float* bias_hh = (const float*)d_in[6];
    float* out = (float*)d_out;

    // 32 subunits x 4 batch tiles = 128 single-wave blocks (spread over WGPs)
    dim3 grid(SUB_NO * (BATCH / 16));
    dim3 block(32);
    sub_mgu_kernel<<<grid, block, 0, stream>>>(x, W_hif, W_hin, W_hhf, W_hhn,
                                               bias_hi, bias_hh, out);
    (void)in_sizes; (void)n_in; (void)out_size; (void)d_ws; (void)ws_size;
}